// B6Model_18502719111544
// MI455X (gfx1250) — compile-verified
//
#include <hip/hip_runtime.h>

#define N_NODES 100000
#define N_EDGES 1600000
#define DIN     67
#define DP      68          // padded feature dim for layer-1 propagation / GEMM K
#define DH      128
#define EPS_BN  1e-5f
#define SLOPE   0.01f

typedef float v2f __attribute__((ext_vector_type(2)));
typedef float v8f __attribute__((ext_vector_type(8)));

// ---------------- utility kernels ----------------

__global__ void zero_kernel(float* __restrict__ p, int n) {
    int i = blockIdx.x * 256 + threadIdx.x;
    if (i < n) p[i] = 0.0f;
}

// degree over target index (col): deg[col[e]] += 1
__global__ void deg_kernel(const int* __restrict__ col, float* __restrict__ deg) {
    int e = blockIdx.x * 256 + threadIdx.x;
    if (e < N_EDGES) atomicAdd(&deg[col[e]], 1.0f);
}

// deg -> dis in place: deg>0 ? rsqrt(max(deg,1)) : 0
__global__ void dis_kernel(float* __restrict__ d) {
    int n = blockIdx.x * 256 + threadIdx.x;
    if (n < N_NODES) {
        float dg = d[n];
        d[n] = (dg > 0.0f) ? rsqrtf(fmaxf(dg, 1.0f)) : 0.0f;
    }
}

__global__ void ew_kernel(const int* __restrict__ row, const int* __restrict__ col,
                          const float* __restrict__ dis, float* __restrict__ ew) {
    int e = blockIdx.x * 256 + threadIdx.x;
    if (e < N_EDGES) ew[e] = dis[row[e]] * dis[col[e]];
}

// BatchNorm1d(67) eval -> H0 padded to 68 cols (pad col = 0)
__global__ void bn1_kernel(const float* __restrict__ x,
                           const float* __restrict__ g, const float* __restrict__ b,
                           const float* __restrict__ m, const float* __restrict__ v,
                           float* __restrict__ h0) {
    int i = blockIdx.x * 256 + threadIdx.x;
    if (i >= N_NODES * DP) return;
    int n = i / DP, d = i - n * DP;
    float val = 0.0f;
    if (d < DIN) {
        float xv = x[n * DIN + d];
        val = (xv - m[d]) * rsqrtf(v[d] + EPS_BN) * g[d] + b[d];
    }
    h0[i] = val;
}

// one normalized propagation hop at padded dim 68, float4-vectorized scatter-add
__global__ void hop68_kernel(const int* __restrict__ row, const int* __restrict__ col,
                             const float* __restrict__ ew,
                             const float* __restrict__ hin, float* __restrict__ hout) {
    int t = blockIdx.x * 256 + threadIdx.x;          // E * 17 threads
    if (t >= N_EDGES * 17) return;
    int e = t / 17;
    int f = (t - e * 17) * 4;
    float w = ew[e];
    int r = row[e], c = col[e];
    const float4 hv = *(const float4*)(hin + r * DP + f);
    float* dst = hout + c * DP + f;
    atomicAdd(dst + 0, hv.x * w);
    atomicAdd(dst + 1, hv.y * w);
    atomicAdd(dst + 2, hv.z * w);
    atomicAdd(dst + 3, hv.w * w);
}

// pack W1[4][128][67] -> Wt[(k*68+kk)*128 + o], K padded with zeros
__global__ void packwt_kernel(const float* __restrict__ W1, float* __restrict__ Wt) {
    int i = blockIdx.x * 256 + threadIdx.x;          // 4*68*128 threads
    if (i >= 4 * DP * DH) return;
    int o  = i & (DH - 1);
    int kr = i >> 7;
    int k  = kr / DP;
    int kk = kr - k * DP;
    Wt[i] = (kk < DIN) ? W1[(k * DH + o) * DIN + kk] : 0.0f;
}

// fused: out1 = sum_k Hk @ Wtk + bias1 ; BN2 ; leaky_relu  -> hp [N x 128]
// one wave per 16x16 tile, V_WMMA_F32_16X16X4_F32
__global__ void gemm_wmma_kernel(const float* __restrict__ H, const float* __restrict__ Wt,
                                 const float* __restrict__ bias1,
                                 const float* __restrict__ g2, const float* __restrict__ b2,
                                 const float* __restrict__ m2, const float* __restrict__ v2,
                                 float* __restrict__ hp) {
    int wave  = (blockIdx.x << 2) + (threadIdx.x >> 5);  // 50000 waves total
    int lane  = threadIdx.x & 31;
    int tileN = wave & 7;
    int tileM = wave >> 3;
    int laneHalf = lane >> 4;      // 0/1
    int lane15   = lane & 15;

    int aRow = tileM * 16 + lane15;
    int bCol = tileN * 16 + lane15;

    v8f c = {};
    for (int hop = 0; hop < 4; ++hop) {
        const float* Hk = H  + (size_t)hop * (size_t)(N_NODES * DP) + (size_t)aRow * DP + 2 * laneHalf;
        const float* Wk = Wt + (size_t)hop * (DP * DH) + (2 * laneHalf) * DH + bCol;
        #pragma unroll
        for (int k0 = 0; k0 < DP; k0 += 4) {
            // A fragment: 16x4 f32 -> lane holds (row=aRow, K = k0+2*laneHalf .. +1)
            v2f a = *(const v2f*)(Hk + k0);
            // B fragment: 4x16 f32 -> lane holds (K = k0+2*laneHalf .. +1, col=bCol)
            v2f b;
            b.x = Wk[k0 * DH];
            b.y = Wk[k0 * DH + DH];
            c = __builtin_amdgcn_wmma_f32_16x16x4_f32(false, a, false, b,
                                                      (short)0, c, false, false);
        }
    }
    // epilogue: bias1 + BN2 + leaky_relu, store hp
    float bc = bias1[bCol];
    float sc = rsqrtf(v2[bCol] + EPS_BN) * g2[bCol];
    float mm = m2[bCol];
    float bb = b2[bCol];
    #pragma unroll
    for (int r = 0; r < 8; ++r) {
        int rowo = tileM * 16 + r + 8 * laneHalf;   // C/D layout: VGPR r, lanes>=16 -> M=r+8
        float val = c[r] + bc;
        val = (val - mm) * sc + bb;
        val = (val > 0.0f) ? val : SLOPE * val;
        hp[rowo * DH + bCol] = val;
    }
}

// z_k[n] = sum_d hp[n,d] * W2[k,0,d]   (one wave per node)
__global__ void zk_kernel(const float* __restrict__ hp, const float* __restrict__ W2,
                          float* __restrict__ z) {
    int n    = (blockIdx.x << 3) + (threadIdx.x >> 5);   // 8 waves / block, exact
    int lane = threadIdx.x & 31;
    const float* h = hp + (size_t)n * DH;
    float h0 = h[lane], h1 = h[lane + 32], h2 = h[lane + 64], h3 = h[lane + 96];
    #pragma unroll
    for (int k = 0; k < 4; ++k) {
        const float* w = W2 + k * DH;
        float acc = h0 * w[lane] + h1 * w[lane + 32] + h2 * w[lane + 64] + h3 * w[lane + 96];
        for (int off = 16; off > 0; off >>= 1) acc += __shfl_down(acc, off);
        if (lane == 0) z[k * N_NODES + n] = acc;
    }
}

// scalar propagation hop: v[col] += u[row]*ew
__global__ void shop_kernel(const int* __restrict__ row, const int* __restrict__ col,
                            const float* __restrict__ ew,
                            const float* __restrict__ u, float* __restrict__ v) {
    int e = blockIdx.x * 256 + threadIdx.x;
    if (e < N_EDGES) atomicAdd(&v[col[e]], u[row[e]] * ew[e]);
}

__global__ void add_kernel(float* __restrict__ a, const float* __restrict__ b) {
    int n = blockIdx.x * 256 + threadIdx.x;
    if (n < N_NODES) a[n] += b[n];
}

__global__ void final_kernel(const float* __restrict__ s, const float* __restrict__ z0,
                             const float* __restrict__ bias2, float* __restrict__ out) {
    int n = blockIdx.x * 256 + threadIdx.x;
    if (n < N_NODES) out[n] = s[n] + z0[n] + bias2[0];
}

// ---------------- launch ----------------

extern "C" void kernel_launch(void* const* d_in, const int* in_sizes, int n_in,
                              void* d_out, int out_size, void* d_ws, size_t ws_size,
                              hipStream_t stream) {
    const float* x     = (const float*)d_in[0];
    const int*   ei    = (const int*)  d_in[1];
    const float* g1    = (const float*)d_in[2];
    const float* b1    = (const float*)d_in[3];
    const float* m1    = (const float*)d_in[4];
    const float* v1    = (const float*)d_in[5];
    const float* W1    = (const float*)d_in[6];
    const float* bias1 = (const float*)d_in[7];
    const float* g2    = (const float*)d_in[8];
    const float* b2    = (const float*)d_in[9];
    const float* m2    = (const float*)d_in[10];
    const float* v2    = (const float*)d_in[11];
    const float* W2    = (const float*)d_in[12];
    const float* bias2 = (const float*)d_in[13];
    float* out = (float*)d_out;

    const int* row = ei;            // edge_index[0]
    const int* col = ei + N_EDGES;  // edge_index[1]

    // workspace layout (floats)
    float* ws = (float*)d_ws;
    size_t off = 0;
    auto alloc = [&](size_t n) { float* p = ws + off; off += (n + 63) & ~(size_t)63; return p; };
    float* dis = alloc(N_NODES);                 // deg, then dis in place
    float* ew  = alloc(N_EDGES);
    float* H   = alloc((size_t)4 * N_NODES * DP);// H0..H3 contiguous
    float* Wt  = alloc(4 * DP * DH);
    float* hp  = alloc((size_t)N_NODES * DH);
    float* z   = alloc((size_t)4 * N_NODES);     // z0..z3
    float* sA  = alloc(N_NODES);
    float* sB  = alloc(N_NODES);

    const int BN   = 256;
    const int gN   = (N_NODES + BN - 1) / BN;
    const int gE   = (N_EDGES + BN - 1) / BN;
    const int gBN1 = (N_NODES * DP + BN - 1) / BN;
    const int gZH  = (3 * N_NODES * DP + BN - 1) / BN;
    const int gHop = (N_EDGES * 17 + BN - 1) / BN;

    // edge normalization
    zero_kernel<<<gN, BN, 0, stream>>>(dis, N_NODES);
    deg_kernel <<<gE, BN, 0, stream>>>(col, dis);
    dis_kernel <<<gN, BN, 0, stream>>>(dis);
    ew_kernel  <<<gE, BN, 0, stream>>>(row, col, dis, ew);

    // layer 1: BN -> 3 propagation hops at d=68
    bn1_kernel <<<gBN1, BN, 0, stream>>>(x, g1, b1, m1, v1, H);
    zero_kernel<<<gZH, BN, 0, stream>>>(H + (size_t)N_NODES * DP, 3 * N_NODES * DP);
    hop68_kernel<<<gHop, BN, 0, stream>>>(row, col, ew, H,                              H + (size_t)N_NODES * DP);
    hop68_kernel<<<gHop, BN, 0, stream>>>(row, col, ew, H + (size_t)N_NODES * DP,       H + (size_t)2 * N_NODES * DP);
    hop68_kernel<<<gHop, BN, 0, stream>>>(row, col, ew, H + (size_t)2 * N_NODES * DP,   H + (size_t)3 * N_NODES * DP);

    // fused WMMA GEMM (+bias1 +BN2 +leaky) -> hp
    packwt_kernel<<<(4 * DP * DH) / BN, BN, 0, stream>>>(W1, Wt);
    gemm_wmma_kernel<<<(N_NODES / 16) * 8 / 4, 128, 0, stream>>>(H, Wt, bias1, g2, b2, m2, v2, hp);

    // layer 2 via commuted scalar propagation: out = z0 + A(z1 + A(z2 + A z3)) + bias2
    zk_kernel<<<N_NODES / 8, 256, 0, stream>>>(hp, W2, z);

    zero_kernel<<<gN, BN, 0, stream>>>(sB, N_NODES);
    shop_kernel<<<gE, BN, 0, stream>>>(row, col, ew, z + 3 * N_NODES, sB);
    add_kernel <<<gN, BN, 0, stream>>>(sB, z + 2 * N_NODES);

    zero_kernel<<<gN, BN, 0, stream>>>(sA, N_NODES);
    shop_kernel<<<gE, BN, 0, stream>>>(row, col, ew, sB, sA);
    add_kernel <<<gN, BN, 0, stream>>>(sA, z + 1 * N_NODES);

    zero_kernel<<<gN, BN, 0, stream>>>(sB, N_NODES);
    shop_kernel<<<gE, BN, 0, stream>>>(row, col, ew, sA, sB);
    final_kernel<<<gN, BN, 0, stream>>>(sB, z, bias2, out);
}